// Point_context_produce_6047313953083
// MI455X (gfx1250) — compile-verified
//
#include <hip/hip_runtime.h>
#include <math.h>

// ---------------- problem sizes (match reference) ----------------
#define N1   32768
#define N2   32768
#define N1D  4096
#define N2D  4096
#define CCH  64
#define M3C  192   // 3*C

typedef __attribute__((ext_vector_type(2))) float v2f;
typedef __attribute__((ext_vector_type(4))) float v4f;
typedef __attribute__((ext_vector_type(8))) float v8f;
typedef int v4i __attribute__((vector_size(16)));

// address-space-qualified pointer element types for the async builtin
typedef __attribute__((address_space(1))) v4i gas_v4i;   // global
typedef __attribute__((address_space(3))) v4i las_v4i;   // LDS

// ---- CDNA5 async global->LDS copy (ASYNCcnt-tracked), with safe fallback ----
#if defined(__has_builtin)
#if __has_builtin(__builtin_amdgcn_global_load_async_to_lds_b128)
#define HAVE_ASYNC_LDS 1
#endif
#endif
#ifndef HAVE_ASYNC_LDS
#define HAVE_ASYNC_LDS 0
#endif

__device__ __forceinline__ void async_copy16(float* lds_dst, const float* gsrc) {
#if HAVE_ASYNC_LDS
  __builtin_amdgcn_global_load_async_to_lds_b128(
      (gas_v4i*)gsrc, (las_v4i*)lds_dst, 0, 0);
#else
  *(v4f*)lds_dst = *(const v4f*)gsrc;
#endif
}

__device__ __forceinline__ void wait_async_done() {
#if HAVE_ASYNC_LDS
#if __has_builtin(__builtin_amdgcn_s_wait_asynccnt)
  __builtin_amdgcn_s_wait_asynccnt(0);
#else
  asm volatile("s_wait_asynccnt 0x0" ::: "memory");
#endif
#endif
}

// cooperative block copy: nq float4 quads, blockDim.x == 256
__device__ __forceinline__ void block_copy16(float* lds_dst, const float* gsrc,
                                             int nq, int tid) {
  for (int i = tid; i < nq; i += 256)
    async_copy16(lds_dst + (size_t)i * 4, gsrc + (size_t)i * 4);
}

// D = A(16x4) * B(4x16) + C, fp32 WMMA (CDNA5 V_WMMA_F32_16X16X4_F32)
__device__ __forceinline__ v8f wmma_f32_16x16x4(v2f a, v2f b, v8f c) {
  return __builtin_amdgcn_wmma_f32_16x16x4_f32(
      false, a, false, b, (short)0, c, false, false);
}

// branchless insert into running sorted top-3 (d0<=d1<=d2), with indices
__device__ __forceinline__ void insert3(float d, int i,
                                        float& d0, float& d1, float& d2,
                                        int& i0, int& i1, int& i2) {
  bool c0 = d < d0, c1 = d < d1, c2 = d < d2;
  float nd2 = c1 ? d1 : (c2 ? d : d2);
  int   ni2 = c1 ? i1 : (c2 ? i : i2);
  float nd1 = c0 ? d0 : (c1 ? d : d1);
  int   ni1 = c0 ? i0 : (c1 ? i : i1);
  float nd0 = c0 ? d  : d0;
  int   ni0 = c0 ? i  : i0;
  d0 = nd0; d1 = nd1; d2 = nd2;
  i0 = ni0; i1 = ni1; i2 = ni2;
}

// ---------------- pack refs for WMMA A-matrix: (-2x,-2y,-2z, |r|^2) ----------
__global__ void pack_refs_kernel(const float* __restrict__ xyz,
                                 float* __restrict__ packed, int n) {
  int i = blockIdx.x * blockDim.x + threadIdx.x;
  if (i < n) {
    float x = xyz[i * 3 + 0], y = xyz[i * 3 + 1], z = xyz[i * 3 + 2];
    packed[i * 4 + 0] = -2.0f * x;
    packed[i * 4 + 1] = -2.0f * y;
    packed[i * 4 + 2] = -2.0f * z;
    packed[i * 4 + 3] = x * x + y * y + z * z;
  }
}

// ---------------- GEMM: out[M,N] = act(A[M,K] @ W[K,N] + b[N]) ---------------
__global__ void gemm_bias_kernel(const float* __restrict__ A,
                                 const float* __restrict__ W,
                                 const float* __restrict__ bias,
                                 float* __restrict__ out,
                                 int M, int N, int Kdim, int do_relu) {
  const int lane = threadIdx.x & 31;
  const int tile = blockIdx.x * (blockDim.x >> 5) + (threadIdx.x >> 5);
  const int ntn  = N >> 4;
  const int tm = tile / ntn, tn = tile % ntn;
  if (tm >= (M >> 4)) return;              // wave-uniform exit
  const int m0 = tm * 16, n0 = tn * 16;
  const int l15 = lane & 15;
  const int hi  = lane >> 4;

  v8f acc = {};
  const float* arow = A + (size_t)(m0 + l15) * Kdim + 2 * hi;
  const float* wcol = W + (size_t)(2 * hi) * N + n0 + l15;
  for (int k = 0; k < Kdim; k += 4) {
    v2f a, b;
    a.x = arow[0];  a.y = arow[1];
    b.x = wcol[0];  b.y = wcol[N];
    acc = wmma_f32_16x16x4(a, b, acc);
    arow += 4;
    wcol += (size_t)4 * N;
  }
  const float bb = bias[n0 + l15];
#pragma unroll
  for (int v = 0; v < 8; ++v) {
    float val = acc[v] + bb;
    if (do_relu) val = fmaxf(val, 0.0f);
    out[(size_t)(m0 + v + hi * 8) * N + n0 + l15] = val;
  }
}

// ---------------- KNN1 + inverse-distance feature interpolation --------------
// one wave per 16 queries; ref table staged through LDS in double-buffered
// 32KB chunks via async global->LDS DMA, shared by the block's 8 waves.
#define K1_CREFS  2048                 // refs per chunk (32 KB)
#define K1_CTILES (K1_CREFS / 16)      // 128 WMMA tiles per chunk
#define K1_NCHUNK (N1 / K1_CREFS)      // 16 chunks

__global__ void __launch_bounds__(256)
knn1_interp_kernel(const float* __restrict__ packed,  // [N1,4]
                   const float* __restrict__ qxyz,    // [N2,3]
                   const float* __restrict__ feat,    // [N1,C]
                   float* __restrict__ out) {         // [N2,C]
  __shared__ float lbuf[2][K1_CREFS * 4];             // 2 x 32 KB
  const int tid  = threadIdx.x;
  const int lane = tid & 31;
  const int tile = blockIdx.x * 8 + (tid >> 5);
  const int l15  = lane & 15;
  const int hi   = lane >> 4;
  const int q    = tile * 16 + l15;

  const float qx = qxyz[q * 3 + 0];
  const float qy = qxyz[q * 3 + 1];
  const float qz = qxyz[q * 3 + 2];
  v2f bfrag;
  bfrag.x = hi ? qz : qx;
  bfrag.y = hi ? 1.0f : qy;

  float d0 = 3.0e38f, d1 = 3.0e38f, d2 = 3.0e38f;
  int   i0 = 0, i1 = 0, i2 = 0;

  // prologue: stage chunk 0
  block_copy16(lbuf[0], packed, K1_CREFS, tid);
  wait_async_done();
  __syncthreads();

  for (int c = 0; c < K1_NCHUNK; ++c) {
    if (c + 1 < K1_NCHUNK)   // prefetch next chunk while computing this one
      block_copy16(lbuf[(c + 1) & 1],
                   packed + (size_t)(c + 1) * K1_CREFS * 4, K1_CREFS, tid);

    const float* lp = lbuf[c & 1] + l15 * 4 + 2 * hi;
    const int rbase = c * K1_CREFS + hi * 8;
#pragma unroll 4
    for (int t = 0; t < K1_CTILES; ++t) {
      v2f a; a.x = lp[0]; a.y = lp[1];
      v8f acc = {};
      acc = wmma_f32_16x16x4(a, bfrag, acc);
      const int rb = rbase + t * 16;
#pragma unroll
      for (int v = 0; v < 8; ++v) insert3(acc[v], rb + v, d0, d1, d2, i0, i1, i2);
      lp += 64;  // 16 refs * 4 floats
    }
    wait_async_done();
    __syncthreads();
  }

  // merge the two half-wave candidate sets (ref rows 0-7 vs 8-15)
  float e0 = __shfl_xor(d0, 16, 32), e1 = __shfl_xor(d1, 16, 32), e2 = __shfl_xor(d2, 16, 32);
  int   j0 = __shfl_xor(i0, 16, 32), j1 = __shfl_xor(i1, 16, 32), j2 = __shfl_xor(i2, 16, 32);
  insert3(e0, j0, d0, d1, d2, i0, i1, i2);
  insert3(e1, j1, d0, d1, d2, i0, i1, i2);
  insert3(e2, j2, d0, d1, d2, i0, i1, i2);
  // every lane now holds the top-3 of query (tile*16 + l15)

  // exact weights (reference: dist = max(norm, 1e-10); w = 1/dist; normalize)
  int idx[3] = {i0, i1, i2};
  float w[3]; float wsum = 0.0f;
#pragma unroll
  for (int k = 0; k < 3; ++k) {
    const float* pp = packed + (size_t)idx[k] * 4;
    float rx = -0.5f * pp[0], ry = -0.5f * pp[1], rz = -0.5f * pp[2];
    float dx = rx - qx, dy = ry - qy, dz = rz - qz;
    float dist = sqrtf(dx * dx + dy * dy + dz * dz);
    dist = fmaxf(dist, 1e-10f);
    w[k] = 1.0f / dist;
    wsum += w[k];
  }
  const float inv = 1.0f / wsum;
  w[0] *= inv; w[1] *= inv; w[2] *= inv;

  // coalesced interpolation: per query, all 32 lanes sweep the 64 channels
  for (int qq = 0; qq < 16; ++qq) {
    float bw0 = __shfl(w[0], qq, 32);
    float bw1 = __shfl(w[1], qq, 32);
    float bw2 = __shfl(w[2], qq, 32);
    int   bi0 = __shfl(idx[0], qq, 32);
    int   bi1 = __shfl(idx[1], qq, 32);
    int   bi2 = __shfl(idx[2], qq, 32);
    const float* f0 = feat + (size_t)bi0 * CCH;
    const float* f1 = feat + (size_t)bi1 * CCH;
    const float* f2 = feat + (size_t)bi2 * CCH;
    float* o = out + (size_t)(tile * 16 + qq) * CCH;
    o[lane]      = bw0 * f0[lane]      + bw1 * f1[lane]      + bw2 * f2[lane];
    o[lane + 32] = bw0 * f0[lane + 32] + bw1 * f1[lane + 32] + bw2 * f2[lane + 32];
  }
}

// ---------------- KNN2 (per-channel displaced queries) + warp ----------------
// entire 64 KB packed ref table staged once per block into LDS.
__global__ void __launch_bounds__(256)
knn2_warp_kernel(const float* __restrict__ packed2,   // [N1D,4]
                 const float* __restrict__ xyz2d,     // [N2D,3]
                 const float* __restrict__ mbuf,      // [N2D,3C]
                 const float* __restrict__ point1,    // [N1D,C]
                 float* __restrict__ out) {           // [N2D*C]
  __shared__ float lbuf[N1D * 4];                     // 64 KB
  const int tid  = threadIdx.x;
  const int lane = tid & 31;
  const int tile = blockIdx.x * 8 + (tid >> 5);
  const int l15  = lane & 15;
  const int hi   = lane >> 4;
  const int qi = tile * 16 + l15;          // flat query id in [0, N2D*C)
  const int np = qi >> 6;                  // point index (C == 64)
  const int ch = qi & 63;                  // channel

  block_copy16(lbuf, packed2, N1D, tid);

  const float* mv = mbuf + (size_t)np * M3C + ch * 3;
  const float qx = xyz2d[np * 3 + 0] + mv[0];
  const float qy = xyz2d[np * 3 + 1] + mv[1];
  const float qz = xyz2d[np * 3 + 2] + mv[2];
  v2f bfrag;
  bfrag.x = hi ? qz : qx;
  bfrag.y = hi ? 1.0f : qy;

  wait_async_done();
  __syncthreads();

  float d0 = 3.0e38f, d1 = 3.0e38f, d2 = 3.0e38f;
  int   i0 = 0, i1 = 0, i2 = 0;

  const float* lp = lbuf + l15 * 4 + 2 * hi;
#pragma unroll 4
  for (int t = 0; t < (N1D >> 4); ++t) {
    v2f a; a.x = lp[0]; a.y = lp[1];
    v8f acc = {};
    acc = wmma_f32_16x16x4(a, bfrag, acc);
    const int rb = t * 16 + hi * 8;
#pragma unroll
    for (int v = 0; v < 8; ++v) insert3(acc[v], rb + v, d0, d1, d2, i0, i1, i2);
    lp += 64;
  }

  float e0 = __shfl_xor(d0, 16, 32), e1 = __shfl_xor(d1, 16, 32), e2 = __shfl_xor(d2, 16, 32);
  int   j0 = __shfl_xor(i0, 16, 32), j1 = __shfl_xor(i1, 16, 32), j2 = __shfl_xor(i2, 16, 32);
  insert3(e0, j0, d0, d1, d2, i0, i1, i2);
  insert3(e1, j1, d0, d1, d2, i0, i1, i2);
  insert3(e2, j2, d0, d1, d2, i0, i1, i2);

  if (hi == 0) {   // after all WMMAs; one lane per query writes the output
    int idx[3] = {i0, i1, i2};
    float w[3]; float wsum = 0.0f; float acc = 0.0f;
#pragma unroll
    for (int k = 0; k < 3; ++k) {
      const float* pp = lbuf + (size_t)idx[k] * 4;   // coords from LDS
      float rx = -0.5f * pp[0], ry = -0.5f * pp[1], rz = -0.5f * pp[2];
      float dx = rx - qx, dy = ry - qy, dz = rz - qz;
      float sq = dx * dx + dy * dy + dz * dz + 1e-8f;   // reference: d2 + 1e-8
      w[k] = 1.0f / sq;
      wsum += w[k];
    }
    const float inv = 1.0f / fmaxf(wsum, 3.0f);          // reference clamp
#pragma unroll
    for (int k = 0; k < 3; ++k)
      acc += w[k] * point1[(size_t)idx[k] * CCH + ch];
    out[qi] = acc * inv;
  }
}

// ---------------- host-side launch -------------------------------------------
extern "C" void kernel_launch(void* const* d_in, const int* in_sizes, int n_in,
                              void* d_out, int out_size, void* d_ws, size_t ws_size,
                              hipStream_t stream) {
  const float* xyz1      = (const float*)d_in[0];
  const float* f1_feat   = (const float*)d_in[1];
  const float* xyz2      = (const float*)d_in[2];
  const float* xyz1_down = (const float*)d_in[3];
  const float* f1_down   = (const float*)d_in[4];
  const float* xyz2_down = (const float*)d_in[5];
  const float* motion    = (const float*)d_in[6];
  const float* W_f1      = (const float*)d_in[7];
  const float* b_f1      = (const float*)d_in[8];
  const float* W_m1      = (const float*)d_in[9];
  const float* b_m1      = (const float*)d_in[10];
  const float* W_m2      = (const float*)d_in[11];
  const float* b_m2      = (const float*)d_in[12];

  float* out    = (float*)d_out;
  float* warped = out;                       // [N2D*C] (first tuple element)
  float* newf1  = out + (size_t)N2D * CCH;   // [N2*C]

  float* ws      = (float*)d_ws;
  float* packed1 = ws;                          // N1*4
  float* packed2 = packed1 + (size_t)N1 * 4;    // N1D*4
  float* hbuf    = packed2 + (size_t)N1D * 4;   // N2D*3C
  float* mbuf    = hbuf + (size_t)N2D * M3C;    // N2D*3C
  float* p1buf   = mbuf + (size_t)N2D * M3C;    // N1D*C

  pack_refs_kernel<<<(N1 + 255) / 256, 256, 0, stream>>>(xyz1, packed1, N1);
  pack_refs_kernel<<<(N1D + 255) / 256, 256, 0, stream>>>(xyz1_down, packed2, N1D);

  // motion MLP: relu(motion @ W_m1 + b_m1) @ W_m2 + b_m2
  {
    int tiles = (N2D / 16) * (M3C / 16);
    gemm_bias_kernel<<<tiles / 8, 256, 0, stream>>>(motion, W_m1, b_m1, hbuf,
                                                    N2D, M3C, M3C, 1);
    gemm_bias_kernel<<<tiles / 8, 256, 0, stream>>>(hbuf, W_m2, b_m2, mbuf,
                                                    N2D, M3C, M3C, 0);
  }
  // point1 = f1_down @ W_f1 + b_f1
  {
    int tiles = (N1D / 16) * (CCH / 16);
    gemm_bias_kernel<<<tiles / 8, 256, 0, stream>>>(f1_down, W_f1, b_f1, p1buf,
                                                    N1D, CCH, CCH, 0);
  }

  // KNN1 + interpolation (2048 query tiles, 8 waves/block)
  knn1_interp_kernel<<<(N2 / 16) / 8, 256, 0, stream>>>(packed1, xyz2, f1_feat, newf1);

  // KNN2 + per-channel warp (16384 query tiles)
  knn2_warp_kernel<<<((N2D * CCH) / 16) / 8, 256, 0, stream>>>(packed2, xyz2_down,
                                                               mbuf, p1buf, warped);
}